// bb_creation_v3_43843026157745
// MI455X (gfx1250) — compile-verified
//
#include <hip/hip_runtime.h>
#include <hip/hip_bf16.h>
#include <cstdint>

// Problem constants (from the reference)
#define N_CLASSES   81
#define N_ROIS      32
#define N_CM1       80                       // CLASSES - 1
#define OUT_COLS    (N_CLASSES + 4)          // 85
#define OUT_PER_C   (N_ROIS * OUT_COLS)      // 2720 floats per class-slice
#define CLS_ELEMS   (N_ROIS * N_CLASSES)     // 2592 floats of out_class
#define IMG_W       1000.0f                  // b.shape[2]
#define IMG_H       600.0f                   // b.shape[1]

typedef __attribute__((ext_vector_type(4))) unsigned int v4u;
typedef __attribute__((ext_vector_type(8))) int          v8i;
typedef __attribute__((ext_vector_type(4))) int          v4i;

#ifndef __has_builtin
#define __has_builtin(x) 0
#endif

__global__ __launch_bounds__(256) void bb_creation_kernel(
    const float* __restrict__ out_regr,   // (32, 320) = (R, 4*(C-1))
    const float* __restrict__ out_class,  // (32, 81)
    const float* __restrict__ rois,       // (32, 4)
    const int*   __restrict__ masks,      // (80, 32)
    float*       __restrict__ out)        // (80, 32, 85)
{
    __shared__ float s_cls[CLS_ELEMS];    // 10368 B: broadcast class scores
    __shared__ float s_coord[N_ROIS * 4]; //   512 B: per-(c,r) coords
    __shared__ float s_mask[N_ROIS];      //   128 B

    const int c   = blockIdx.x;           // class slice this block produces
    const int tid = threadIdx.x;

#if __has_builtin(__builtin_amdgcn_tensor_load_to_lds)
    // ---- CDNA5 Tensor Data Mover: stage out_class (2592 f32) into LDS ----
    // 1-D descriptor: count=1, data_size=4B (code 2), tensor_dim0 = tile_dim0
    // = 2592 elements, higher dims unused (groups 2/3 zero => <=2D tensor).
    if (tid < 32) {  // wave 0 only (wave-uniform branch); TDM ignores EXEC
        const uint64_t ga  = (uint64_t)(uintptr_t)out_class;       // 57-bit VA
        const uint32_t lds = (uint32_t)(uintptr_t)s_cls;           // flat[31:0] = LDS offset

        v4u g0;
        g0.x = 1u;                                   // count=1, user mode
        g0.y = lds;                                  // lds_addr [63:32]
        g0.z = (uint32_t)ga;                         // global_addr[31:0]
        g0.w = (uint32_t)((ga >> 32) & 0x01FFFFFFu)  // global_addr[56:32]
             | (2u << 30);                           // type = 2 ("image")

        v8i g1;
        g1[0] = (int)(2u << 16);                     // wg_mask=0, data_size=4B
        g1[1] = (int)((unsigned)CLS_ELEMS << 16);    // tensor_dim0[15:0]
        g1[2] = (int)(1u << 16);                     // tensor_dim0 hi=0, tensor_dim1=1
        g1[3] = (int)((unsigned)CLS_ELEMS << 16);    // tensor_dim1 hi=0, tile_dim0
        g1[4] = 0;                                   // tile_dim1=0, tile_dim2=0 (unused)
        g1[5] = CLS_ELEMS;                           // tensor_dim0_stride[31:0]
        g1[6] = 0;
        g1[7] = 0;

        v4i gz = {0, 0, 0, 0};
#if defined(__clang_major__) && (__clang_major__ >= 23)
        v8i gz8 = {0, 0, 0, 0, 0, 0, 0, 0};
        __builtin_amdgcn_tensor_load_to_lds(g0, g1, gz, gz, gz8, 0);
#else
        __builtin_amdgcn_tensor_load_to_lds(g0, g1, gz, gz, 0);
#endif
    }
#else
    // Fallback (host pass / toolchain without the TDM builtin): cooperative load.
    for (int i = tid; i < CLS_ELEMS; i += 256) s_cls[i] = out_class[i];
#endif

    // ---- Box decode for this class slice, overlapped with the TDM DMA ----
    if (tid < N_ROIS) {
        const int r = tid;
        const float x = rois[r * 4 + 0];
        const float y = rois[r * 4 + 1];
        const float w = rois[r * 4 + 2];
        const float h = rois[r * 4 + 3];
        const float cx = x + 0.5f * w;
        const float cy = y + 0.5f * h;

        const int  rb = r * (4 * N_CM1) + c * 4;
        // /STD with STD = {8,8,4,4}: exact power-of-two scaling
        const float t0 = out_regr[rb + 0] * 0.125f;
        const float t1 = out_regr[rb + 1] * 0.125f;
        const float t2 = out_regr[rb + 2] * 0.25f;
        const float t3 = out_regr[rb + 3] * 0.25f;

        const float cx1 = t0 * w + cx;
        const float cy1 = t1 * h + cy;
        const float w1  = expf(t2) * w;
        const float h1  = expf(t3) * h;

        // jnp.round = round-half-to-even = rintf under default RTE mode
        const float gx = rintf(cx1 - 0.5f * w1);
        const float gy = rintf(cy1 - 0.5f * h1);
        const float gw = rintf(w1);
        const float gh = rintf(h1);

        s_coord[r * 4 + 0] = gx / IMG_W;
        s_coord[r * 4 + 1] = gy / IMG_H;
        s_coord[r * 4 + 2] = (gx + gw) / IMG_W;
        s_coord[r * 4 + 3] = (gy + gh) / IMG_H;
        s_mask[r] = (float)masks[c * N_ROIS + r];
    }

#if __has_builtin(__builtin_amdgcn_s_wait_tensorcnt)
    __builtin_amdgcn_s_wait_tensorcnt(0);   // TDM LDS writes visible (no-op for waves with cnt==0)
#endif
    __syncthreads();

    // ---- Stream the 2720 masked outputs of this class slice ----
    float* __restrict__ outc = out + (size_t)c * OUT_PER_C;
    for (int idx = tid; idx < OUT_PER_C; idx += 256) {
        const int r = idx / OUT_COLS;
        const int j = idx - r * OUT_COLS;
        const float v = (j < N_CLASSES) ? s_cls[r * N_CLASSES + j]
                                        : s_coord[r * 4 + (j - N_CLASSES)];
        outc[idx] = v * s_mask[r];
    }
}

extern "C" void kernel_launch(void* const* d_in, const int* in_sizes, int n_in,
                              void* d_out, int out_size, void* d_ws, size_t ws_size,
                              hipStream_t stream) {
    // setup_inputs order: out_regr, out_class, rois, b (unused, shape-only), masks
    const float* out_regr  = (const float*)d_in[0];
    const float* out_class = (const float*)d_in[1];
    const float* rois      = (const float*)d_in[2];
    const int*   masks     = (const int*)  d_in[4];
    float* out = (float*)d_out;

    bb_creation_kernel<<<N_CM1, 256, 0, stream>>>(out_regr, out_class, rois, masks, out);
}